// SpGraphAttentionLayer_81810537054928
// MI455X (gfx1250) — compile-verified
//
#include <hip/hip_runtime.h>
#include <hip/hip_bf16.h>

#define N_NODES 50000
#define N_EDGES 1600000
#define IN_F    128
#define OUT_F   64
#define LRELU_ALPHA 0.2f

typedef float v2f __attribute__((ext_vector_type(2)));
typedef float v8f __attribute__((ext_vector_type(8)));

// W is stored in LDS as K-pairs: sWp[p*PSTR + 2*n + (k&1)] = W[2p + (k&1), n].
// PSTR = 160 floats (mod 64 == 32) -> lane halves (pair-rows p and p+1) hit
// disjoint bank sets for the b64 fragment loads.
#define PSTR 160

// ---------------------------------------------------------------------------
// Kernel 1: h = X @ W   (M x 128) * (128 x 64) with V_WMMA_F32_16X16X4_F32.
// 8 waves / block; each wave computes a 16-row x 64-col strip.
// ---------------------------------------------------------------------------
__launch_bounds__(256)
__global__ void gat_gemm_wmma(const float* __restrict__ X,
                              const float* __restrict__ W,
                              float* __restrict__ h, int M) {
  __shared__ float sWp[(IN_F / 2) * PSTR];   // 40 KB
  const int tid = threadIdx.x;
  // Cooperative load of W (128x64), repacked so each B fragment {W[kk,n],
  // W[kk+1,n]} is one aligned 8-byte LDS load.
  for (int i = tid; i < IN_F * OUT_F; i += 256) {
    int k = i >> 6;
    int n = i & 63;
    sWp[(k >> 1) * PSTR + 2 * n + (k & 1)] = W[i];
  }
  __syncthreads();

  const int wave  = tid >> 5;
  const int lane  = tid & 31;
  const int half  = lane >> 4;   // 0: lanes 0-15, 1: lanes 16-31
  const int l16   = lane & 15;
  const int mBase = blockIdx.x * 128 + wave * 16;

  // A-fragment rows (clamped for the ragged last block; stores are guarded).
  int rowLoad = mBase + l16;
  if (rowLoad > M - 1) rowLoad = M - 1;
  const float* xr = X + (long long)rowLoad * IN_F;

  v8f acc0 = {}, acc1 = {}, acc2 = {}, acc3 = {};

  for (int k0 = 0; k0 < IN_F; k0 += 4) {
    // ISA 32-bit A 16x4 layout: lanes 0-15 hold K=k0,k0+1; lanes 16-31 hold
    // K=k0+2,k0+3 (vgpr0=first, vgpr1=second of the pair).
    const int kk = k0 + 2 * half;
    v2f a = *(const v2f*)(xr + kk);          // 8B-aligned (kk even)

    // B 4x16 fragments: vgpr0 = W[kk, n], vgpr1 = W[kk+1, n] -> one b64 load.
    const float* wp = &sWp[(kk >> 1) * PSTR + 2 * l16];
    v2f b0 = *(const v2f*)(wp +  0);
    v2f b1 = *(const v2f*)(wp + 32);
    v2f b2 = *(const v2f*)(wp + 64);
    v2f b3 = *(const v2f*)(wp + 96);

    acc0 = __builtin_amdgcn_wmma_f32_16x16x4_f32(false, a, false, b0, (short)0, acc0, false, false);
    acc1 = __builtin_amdgcn_wmma_f32_16x16x4_f32(false, a, false, b1, (short)0, acc1, false, false);
    acc2 = __builtin_amdgcn_wmma_f32_16x16x4_f32(false, a, false, b2, (short)0, acc2, false, false);
    acc3 = __builtin_amdgcn_wmma_f32_16x16x4_f32(false, a, false, b3, (short)0, acc3, false, false);
  }

  // C/D layout: element (vgpr j, lane): M = j + 8*half, N = l16 (+ tile base).
#pragma unroll
  for (int j = 0; j < 8; ++j) {
    int gm = mBase + j + 8 * half;
    if (gm < M) {
      float* hr = h + (long long)gm * OUT_F + l16;
      hr[ 0] = acc0[j];
      hr[16] = acc1[j];
      hr[32] = acc2[j];
      hr[48] = acc3[j];
    }
  }
}

// ---------------------------------------------------------------------------
// Kernel 2: per-row attention pre-scores  s_src = h . a[:64], s_dst = h . a[64:]
// ---------------------------------------------------------------------------
__launch_bounds__(256)
__global__ void gat_scores(const float* __restrict__ h,
                           const float* __restrict__ a,
                           float* __restrict__ s_src,
                           float* __restrict__ s_dst, int n) {
  __shared__ float sa[2 * OUT_F];
  if (threadIdx.x < 2 * OUT_F) sa[threadIdx.x] = a[threadIdx.x];
  __syncthreads();

  int r = blockIdx.x * blockDim.x + threadIdx.x;
  if (r >= n) return;
  const float* hr = h + (long long)r * OUT_F;
  float ss = 0.f, sd = 0.f;
#pragma unroll
  for (int i = 0; i < OUT_F; i += 4) {
    float4 v = *(const float4*)(hr + i);
    ss += v.x * sa[i] + v.y * sa[i + 1] + v.z * sa[i + 2] + v.w * sa[i + 3];
    sd += v.x * sa[OUT_F + i] + v.y * sa[OUT_F + i + 1] +
          v.z * sa[OUT_F + i + 2] + v.w * sa[OUT_F + i + 3];
  }
  s_src[r] = ss;
  s_dst[r] = sd;
}

// ---------------------------------------------------------------------------
// Kernel 3: edge scatter. One wave32 per edge: coalesced 128B gathers of
// h[dst], 65 f32 atomics spread 2-per-lane into L2-resident hprime/rowsum.
// ---------------------------------------------------------------------------
__launch_bounds__(256)
__global__ void gat_edges(const long long* __restrict__ esrc,
                          const long long* __restrict__ edst,
                          const float* __restrict__ s_src,
                          const float* __restrict__ s_dst,
                          const float* __restrict__ h,
                          float* __restrict__ hprime,
                          float* __restrict__ rowsum,
                          int nWaves) {
  const int gw   = (blockIdx.x * blockDim.x + threadIdx.x) >> 5;
  const int lane = threadIdx.x & 31;
  for (int e = gw; e < N_EDGES; e += nWaves) {
    const long long i = esrc[e];
    const long long j = edst[e];
    float sc = s_src[i] + s_dst[j];                 // uniform within wave
    float lr = sc > 0.f ? sc : LRELU_ALPHA * sc;
    float w  = __expf(-lr);
    const float* hj = h + j * OUT_F;
    float*       hp = hprime + i * OUT_F;
    atomicAdd(&hp[lane],      w * hj[lane]);
    atomicAdd(&hp[lane + 32], w * hj[lane + 32]);
    if (lane == 0) atomicAdd(&rowsum[i], w);
  }
}

// ---------------------------------------------------------------------------
// Kernel 4: h_prime / rowsum, then ELU.
// ---------------------------------------------------------------------------
__launch_bounds__(256)
__global__ void gat_finalize(const float* __restrict__ hprime,
                             const float* __restrict__ rowsum,
                             float* __restrict__ out, int total) {
  int idx = blockIdx.x * blockDim.x + threadIdx.x;
  if (idx >= total) return;
  int r = idx >> 6;
  float v = hprime[idx] / rowsum[r];
  out[idx] = v > 0.f ? v : (__expf(v) - 1.f);
}

__launch_bounds__(256)
__global__ void gat_zero(float* __restrict__ p, int total) {
  int idx = blockIdx.x * blockDim.x + threadIdx.x;
  if (idx < total) p[idx] = 0.f;
}

// ---------------------------------------------------------------------------
extern "C" void kernel_launch(void* const* d_in, const int* in_sizes, int n_in,
                              void* d_out, int out_size, void* d_ws, size_t ws_size,
                              hipStream_t stream) {
  const float*     X    = (const float*)d_in[0];       // (N, 128) f32
  const long long* edge = (const long long*)d_in[1];   // (2, E)  i64
  const float*     W    = (const float*)d_in[2];       // (128, 64) f32
  const float*     a    = (const float*)d_in[3];       // (1, 128) f32
  float*           out  = (float*)d_out;               // (N, 64) f32
  const int n = N_NODES;

  // Workspace layout: h | hprime | rowsum | s_src | s_dst
  float* h      = (float*)d_ws;
  float* hprime = h      + (size_t)n * OUT_F;
  float* rowsum = hprime + (size_t)n * OUT_F;
  float* s_src  = rowsum + n;
  float* s_dst  = s_src  + n;

  // Zero the accumulation buffers (hprime + rowsum are contiguous).
  {
    int tot = n * (OUT_F + 1);
    gat_zero<<<(tot + 255) / 256, 256, 0, stream>>>(hprime, tot);
  }
  gat_gemm_wmma<<<(n + 127) / 128, 256, 0, stream>>>(X, W, h, n);
  gat_scores<<<(n + 255) / 256, 256, 0, stream>>>(h, a, s_src, s_dst, n);
  {
    const int blocks = 1024;
    const int nWaves = blocks * 256 / 32;
    gat_edges<<<blocks, 256, 0, stream>>>(edge, edge + N_EDGES,
                                          s_src, s_dst, h, hprime, rowsum, nWaves);
  }
  {
    int tot = n * OUT_F;
    gat_finalize<<<(tot + 255) / 256, 256, 0, stream>>>(hprime, rowsum, out, tot);
  }
  (void)in_sizes; (void)n_in; (void)out_size; (void)ws_size;
}